// Layer_66305705115853
// MI455X (gfx1250) — compile-verified
//
#include <hip/hip_runtime.h>
#include <math.h>

// Problem constants (fixed by reference setup_inputs)
constexpr int kB  = 128;
constexpr int kS  = 256;
constexpr int kD  = 128;
constexpr int kH  = 4;
constexpr int kDh = 32;
constexpr int kF  = 512;
constexpr int kNROW = kB * kS;   // 32768

typedef float v2f __attribute__((ext_vector_type(2)));
typedef float v8f __attribute__((ext_vector_type(8)));

__device__ __forceinline__ v8f wmma4(v2f a, v2f b, v8f c) {
  // D = A(16x4 f32) * B(4x16 f32) + C(16x16 f32)
  return __builtin_amdgcn_wmma_f32_16x16x4_f32(false, a, false, b, (short)0, c,
                                               false, false);
}

__device__ __forceinline__ float gelu_f(float v) {
  return 0.5f * v * (1.0f + erff(v * 0.7071067811865475f));
}

// ---------------------------------------------------------------------------
// GEMM: C[M,N] = act(A[M,K] @ W[K,N] + bias[N]); one wave per 16x16 tile.
// grid = (M/16, N/16), block = 32
// ---------------------------------------------------------------------------
__global__ __launch_bounds__(32)
void gemm_bias(const float* __restrict__ A, const float* __restrict__ W,
               const float* __restrict__ bias, float* __restrict__ C,
               int Kd, int N, int act) {
  const int lane = threadIdx.x;
  const int l16  = lane & 15;
  const int half = lane >> 4;            // 0: K=k0,k0+1  1: K=k0+2,k0+3
  const int row  = blockIdx.x * 16 + l16;   // A-operand row (lane = M)
  const int col  = blockIdx.y * 16 + l16;   // B-operand col (lane = N)

  const float* Ar = A + (size_t)row * Kd;
  v8f acc = {0.f, 0.f, 0.f, 0.f, 0.f, 0.f, 0.f, 0.f};

  for (int k0 = 0; k0 < Kd; k0 += 4) {
    const int kk = k0 + 2 * half;
    v2f a; a.x = Ar[kk];                      a.y = Ar[kk + 1];
    v2f b; b.x = W[(size_t)kk * N + col];     b.y = W[(size_t)(kk + 1) * N + col];
    acc = wmma4(a, b, acc);
  }

  const float bc = bias[col];
#pragma unroll
  for (int r = 0; r < 8; ++r) {
    float v = acc[r] + bc;
    if (act == 1) v = gelu_f(v);
    C[((size_t)blockIdx.x * 16 + r + 8 * half) * N + col] = v;  // D: lane=col, r(+8)=row
  }
}

// ---------------------------------------------------------------------------
// Attention: one wave per (b, h, q-tile of 16). S=256, Dh=32.
//   scores^T (k x q) via WMMA, softmax over k (lane-pair reduce), probs in LDS,
//   ctx = P @ V via WMMA.  grid = B*H*(S/16) = 8192, block = 32
// ---------------------------------------------------------------------------
__global__ __launch_bounds__(32)
void attn_kernel(const float* __restrict__ Q, const float* __restrict__ Km,
                 const float* __restrict__ V, const float* __restrict__ mask,
                 float* __restrict__ ctx) {
  const int bid  = blockIdx.x;
  const int qt   = bid & 15;
  const int h    = (bid >> 4) & (kH - 1);
  const int b    = bid >> 6;
  const int lane = threadIdx.x;
  const int l16  = lane & 15;
  const int half = lane >> 4;
  const int q0   = qt * 16;

  __shared__ float sc[kS * 16];            // scores^T: [k][q], 16 KB

  const float scale = 0.17677669529663687f;   // 1/sqrt(32)

  // Preload Q operand chunks: lane l16 = row q, K chunk = 4s + 2*half
  const float* Qb = Q + ((size_t)b * kS + q0 + l16) * kD + h * kDh;
  v2f qreg[8];
#pragma unroll
  for (int s = 0; s < 8; ++s) {
    const int kk = 4 * s + 2 * half;
    qreg[s].x = Qb[kk]; qreg[s].y = Qb[kk + 1];
  }

  const float* Kbase = Km + (size_t)b * kS * kD + h * kDh;
  const float* Mb    = mask + (size_t)b * kS * kS;

  // ---- scores^T = K_tile @ Q^T, scaled + mask, stored to LDS ----
  for (int t = 0; t < 16; ++t) {
    v8f acc = {0.f, 0.f, 0.f, 0.f, 0.f, 0.f, 0.f, 0.f};
    const float* Krow = Kbase + (size_t)(t * 16 + l16) * kD;   // lane = key row
#pragma unroll
    for (int s = 0; s < 8; ++s) {
      const int kk = 4 * s + 2 * half;
      v2f a; a.x = Krow[kk]; a.y = Krow[kk + 1];
      acc = wmma4(a, qreg[s], acc);          // qreg serves as B (Q^T) operand
    }
    const int qg = q0 + l16;
#pragma unroll
    for (int r = 0; r < 8; ++r) {
      const int kg = t * 16 + r + 8 * half;  // D layout: row = r (+8 upper half)
      sc[kg * 16 + l16] = acc[r] * scale + Mb[(size_t)qg * kS + kg];
    }
  }
  __syncthreads();

  // ---- softmax over k per query column; lane pair (j, j+16) splits k ----
  {
    const int kbeg = half * 128;
    float m = -INFINITY;
    for (int k = 0; k < 128; ++k) m = fmaxf(m, sc[(kbeg + k) * 16 + l16]);
    m = fmaxf(m, __shfl_xor(m, 16, 32));
    float sum = 0.f;
    for (int k = 0; k < 128; ++k) {
      const float p = expf(sc[(kbeg + k) * 16 + l16] - m);
      sc[(kbeg + k) * 16 + l16] = p;
      sum += p;
    }
    sum += __shfl_xor(sum, 16, 32);
    const float inv = 1.0f / sum;
    for (int k = 0; k < 128; ++k) sc[(kbeg + k) * 16 + l16] *= inv;
  }
  __syncthreads();

  // ---- ctx[q, d] = sum_k P[q,k] * V[k,d]; two 16-wide d tiles ----
  const float* Vb = V + (size_t)b * kS * kD + h * kDh;
  float* Cb = ctx + ((size_t)b * kS + q0) * kD + h * kDh;
  for (int nt = 0; nt < 2; ++nt) {
    v8f acc = {0.f, 0.f, 0.f, 0.f, 0.f, 0.f, 0.f, 0.f};
    const int d = nt * 16 + l16;
    for (int k0 = 0; k0 < kS; k0 += 4) {
      const int kk = k0 + 2 * half;
      v2f a; a.x = sc[kk * 16 + l16];          a.y = sc[(kk + 1) * 16 + l16]; // P, A-layout
      v2f bb; bb.x = Vb[(size_t)kk * kD + d];  bb.y = Vb[(size_t)(kk + 1) * kD + d];
      acc = wmma4(a, bb, acc);
    }
#pragma unroll
    for (int r = 0; r < 8; ++r)
      Cb[(size_t)(r + 8 * half) * kD + nt * 16 + l16] = acc[r];
  }
}

// ---------------------------------------------------------------------------
// Elementwise row kernels: block = 128 threads (D), grid = B*S rows
// ---------------------------------------------------------------------------
__device__ __forceinline__ float block_reduce_sum128(float v, float* red) {
#pragma unroll
  for (int off = 16; off > 0; off >>= 1) v += __shfl_xor(v, off, 32);
  if ((threadIdx.x & 31) == 0) red[threadIdx.x >> 5] = v;
  __syncthreads();
  const float tot = red[0] + red[1] + red[2] + red[3];
  __syncthreads();
  return tot;
}

__global__ __launch_bounds__(128)
void add_rmsnorm(const float* __restrict__ Hin, const float* __restrict__ X,
                 const float* __restrict__ w, float* __restrict__ out) {
  __shared__ float red[4];
  const size_t i = (size_t)blockIdx.x * kD + threadIdx.x;
  const float t = Hin[i] + X[i];
  const float ss = block_reduce_sum128(t * t, red);
  const float r = rsqrtf(ss / (float)kD + 1e-12f);
  out[i] = w[threadIdx.x] * t * r;
}

__global__ __launch_bounds__(128)
void add_layernorm(const float* __restrict__ Hin, const float* __restrict__ X,
                   const float* __restrict__ w, const float* __restrict__ bi,
                   float* __restrict__ out, int accum) {
  __shared__ float red[4];
  const size_t i = (size_t)blockIdx.x * kD + threadIdx.x;
  const float t = Hin[i] + X[i];
  const float mu = block_reduce_sum128(t, red) / (float)kD;
  const float dv = t - mu;
  const float var = block_reduce_sum128(dv * dv, red) / (float)kD;
  const float o = w[threadIdx.x] * dv * rsqrtf(var + 1e-12f) + bi[threadIdx.x];
  if (accum) out[i] += o; else out[i] = o;
}

// seq_avg (window-5 sum / recursive mask count) + layernorm -> XQ
__global__ __launch_bounds__(128)
void seqavg_ln(const float* __restrict__ X, const float* __restrict__ mask,
               const float* __restrict__ w, const float* __restrict__ bi,
               float* __restrict__ out) {
  __shared__ float red[4];
  const int row = blockIdx.x;
  const int b = row >> 8;
  const int s = row & (kS - 1);
  const int d = threadIdx.x;

  float sum = 0.f;
#pragma unroll
  for (int l = 0; l < 5; ++l)
    if (l <= s) sum += X[((size_t)b * kS + (s - l)) * kD + d];

  // closed form of the recursive mask accumulation: weights = subset-sum
  // multiplicities of {1,2,3,4} -> {1,1,1,2,2,2,2,2,1,1,1} over offsets 0..10
  const float wgt[11] = {1.f,1.f,1.f,2.f,2.f,2.f,2.f,2.f,1.f,1.f,1.f};
  float mavg = 0.f;
#pragma unroll
  for (int dd = 0; dd < 11; ++dd) {
    if (dd <= s) {
      const float mv = mask[(size_t)b * kS * kS + (s - dd)];   // mask[b,0,0,s-dd]
      mavg += (mv > -10000.0f ? 1.0f : 0.0f) * wgt[dd];
    }
  }
  const float u = sum / (mavg + 1e-12f);

  const float mu = block_reduce_sum128(u, red) / (float)kD;
  const float dv = u - mu;
  const float var = block_reduce_sum128(dv * dv, red) / (float)kD;
  out[(size_t)row * kD + d] = w[d] * dv * rsqrtf(var + 1e-12f) + bi[d];
}

// ---------------------------------------------------------------------------
extern "C" void kernel_launch(void* const* d_in, const int* in_sizes, int n_in,
                              void* d_out, int out_size, void* d_ws, size_t ws_size,
                              hipStream_t stream) {
  const float* x    = (const float*)d_in[0];
  const float* mask = (const float*)d_in[1];
  // attn params (no ln): wq bq wk bk wv bv wo bo rms_w
  const float* Awq = (const float*)d_in[2];  const float* Abq = (const float*)d_in[3];
  const float* Awk = (const float*)d_in[4];  const float* Abk = (const float*)d_in[5];
  const float* Awv = (const float*)d_in[6];  const float* Abv = (const float*)d_in[7];
  const float* Awo = (const float*)d_in[8];  const float* Abo = (const float*)d_in[9];
  const float* Arms = (const float*)d_in[10];
  // long params (with ln)
  const float* Lwq = (const float*)d_in[11]; const float* Lbq = (const float*)d_in[12];
  const float* Lwk = (const float*)d_in[13]; const float* Lbk = (const float*)d_in[14];
  const float* Lwv = (const float*)d_in[15]; const float* Lbv = (const float*)d_in[16];
  const float* Lwo = (const float*)d_in[17]; const float* Lbo = (const float*)d_in[18];
  const float* Lrms = (const float*)d_in[19];
  const float* Llnw = (const float*)d_in[20]; const float* Llnb = (const float*)d_in[21];
  // ffn1 / ffn2: w1 b1 w2 b2 ln_w ln_b
  const float* F1w1 = (const float*)d_in[22]; const float* F1b1 = (const float*)d_in[23];
  const float* F1w2 = (const float*)d_in[24]; const float* F1b2 = (const float*)d_in[25];
  const float* F1lw = (const float*)d_in[26]; const float* F1lb = (const float*)d_in[27];
  const float* F2w1 = (const float*)d_in[28]; const float* F2b1 = (const float*)d_in[29];
  const float* F2w2 = (const float*)d_in[30]; const float* F2b2 = (const float*)d_in[31];
  const float* F2lw = (const float*)d_in[32]; const float* F2lb = (const float*)d_in[33];

  float* out = (float*)d_out;
  const size_t RD = (size_t)kNROW * kD;     // 4.19M floats per [B,S,D] buffer
  float* R0  = (float*)d_ws;                // Q   / H scratch
  float* R1  = R0 + RD;                     // K   / branch residual A
  float* R2  = R1 + RD;                     // V
  float* R3  = R2 + RD;                     // ctx
  float* R4  = R3 + RD;                     // XQ (seq_avg_norm)
  float* R5  = R4 + RD;                     // FFN mid [B,S,F]  (16M floats)

  dim3 blk(32);
  dim3 gP(kNROW / 16, kD / 16);             // 2048 x 8
  dim3 gF1(kNROW / 16, kF / 16);            // 2048 x 32
  dim3 gA(kB * kH * (kS / 16));             // 8192
  dim3 eb(128);
  dim3 eg(kNROW);

  // ---------------- branch 1: self_attention + ffn1 ----------------
  gemm_bias<<<gP, blk, 0, stream>>>(x, Awq, Abq, R0, kD, kD, 0);
  gemm_bias<<<gP, blk, 0, stream>>>(x, Awk, Abk, R1, kD, kD, 0);
  gemm_bias<<<gP, blk, 0, stream>>>(x, Awv, Abv, R2, kD, kD, 0);
  attn_kernel<<<gA, blk, 0, stream>>>(R0, R1, R2, mask, R3);
  gemm_bias<<<gP, blk, 0, stream>>>(R3, Awo, Abo, R0, kD, kD, 0);
  add_rmsnorm<<<eg, eb, 0, stream>>>(R0, x, Arms, R1);            // attn_out -> R1
  gemm_bias<<<gF1, blk, 0, stream>>>(R1, F1w1, F1b1, R5, kD, kF, 1);
  gemm_bias<<<gP,  blk, 0, stream>>>(R5, F1w2, F1b2, R0, kF, kD, 0);
  add_layernorm<<<eg, eb, 0, stream>>>(R0, R1, F1lw, F1lb, out, 0);

  // ---------------- branch 2: long_self_attention + ffn2 ----------------
  seqavg_ln<<<eg, eb, 0, stream>>>(x, mask, Llnw, Llnb, R4);       // XQ -> R4
  gemm_bias<<<gP, blk, 0, stream>>>(R4, Lwq, Lbq, R0, kD, kD, 0);
  gemm_bias<<<gP, blk, 0, stream>>>(x,  Lwk, Lbk, R1, kD, kD, 0);
  gemm_bias<<<gP, blk, 0, stream>>>(x,  Lwv, Lbv, R2, kD, kD, 0);
  attn_kernel<<<gA, blk, 0, stream>>>(R0, R1, R2, mask, R3);
  gemm_bias<<<gP, blk, 0, stream>>>(R3, Lwo, Lbo, R0, kD, kD, 0);
  add_rmsnorm<<<eg, eb, 0, stream>>>(R0, R4, Lrms, R1);            // long_out -> R1
  gemm_bias<<<gF1, blk, 0, stream>>>(R1, F2w1, F2b1, R5, kD, kF, 1);
  gemm_bias<<<gP,  blk, 0, stream>>>(R5, F2w2, F2b2, R0, kF, kD, 0);
  add_layernorm<<<eg, eb, 0, stream>>>(R0, R1, F2lw, F2lb, out, 1); // out += out2
}